// PSRoIPool_41042707480837
// MI455X (gfx1250) — compile-verified
//
#include <hip/hip_runtime.h>
#include <cmath>

#define POOLED   7
#define GROUP    7
#define OUT_DIM  21
#define SCALE_F  0.0625f
#define C_       (OUT_DIM * GROUP * GROUP)   // 1029
#define H_       100
#define W_       100
#define OUTS_PER_ROI (OUT_DIM * POOLED * POOLED) // 1029 (== channel count)

// One block per ROI. 256 threads = 8 wave32 waves.
__global__ __launch_bounds__(256) void psroi_pool_kernel(
    const float* __restrict__ feat,   // (B, C_, H_, W_)
    const float* __restrict__ rois,   // (R, 5)
    float* __restrict__ out)          // (R, OUT_DIM, POOLED, POOLED)
{
    __shared__ float s_roi[8];
    __shared__ int   s_hs[POOLED], s_he[POOLED], s_ws[POOLED], s_we[POOLED];

    const int r   = blockIdx.x;
    const int tid = threadIdx.x;

    // ---- CDNA5 async global->LDS staging of the ROI descriptor ----------
    // Shared-aperture flat addresses carry the LDS byte offset in their low
    // 32 bits (aperture base has zero low DWORD per ISA 10.2), and the async
    // op adds LDS_BASE itself, so the truncated pointer is the VDST offset.
    if (tid < 5) {
        unsigned loff = (unsigned)(size_t)(&s_roi[tid]);
        const float* gp = rois + (size_t)r * 5 + tid;
        asm volatile("global_load_async_to_lds_b32 %0, %1, off"
                     :: "v"(loff), "v"((unsigned long long)(size_t)gp)
                     : "memory");
    }
    // Wave 0 waits for its async copies (ASYNCcnt), then block-wide barrier
    // makes the LDS data visible to all 8 waves.
    asm volatile("s_wait_asynccnt 0x0" ::: "memory");
    __syncthreads();

    // ---- per-ROI bin boundaries (mirrors the reference f32 math) --------
    if (tid < POOLED) {
        // jnp.round == round-half-to-even -> rintf under default RNE mode
        float sw = rintf(s_roi[1]) * SCALE_F;
        float sh = rintf(s_roi[2]) * SCALE_F;
        float ew = (rintf(s_roi[3]) + 1.0f) * SCALE_F;
        float eh = (rintf(s_roi[4]) + 1.0f) * SCALE_F;
        float roi_w = fmaxf(ew - sw, 0.1f);
        float roi_h = fmaxf(eh - sh, 0.1f);
        float bin_w = roi_w / (float)POOLED;
        float bin_h = roi_h / (float)POOLED;
        float p = (float)tid;
        s_hs[tid] = (int)fminf(fmaxf(floorf(p * bin_h + sh),          0.0f), (float)H_);
        s_he[tid] = (int)fminf(fmaxf(ceilf((p + 1.0f) * bin_h + sh),  0.0f), (float)H_);
        s_ws[tid] = (int)fminf(fmaxf(floorf(p * bin_w + sw),          0.0f), (float)W_);
        s_we[tid] = (int)fminf(fmaxf(ceilf((p + 1.0f) * bin_w + sw),  0.0f), (float)W_);
    }
    __syncthreads();

    const int b = (int)s_roi[0];
    const float* __restrict__ fb = feat + (size_t)b * C_ * H_ * W_;
    float* __restrict__ ob = out + (size_t)r * OUTS_PER_ROI;

    // Flat local output index o = d*49 + ph*7 + pw == the channel index.
    for (int o = tid; o < OUTS_PER_ROI; o += blockDim.x) {
        int d   = o / (POOLED * POOLED);
        int rem = o - d * (POOLED * POOLED);
        int ph  = rem / POOLED;
        int pw  = rem - ph * POOLED;

        int hs = s_hs[ph], he = s_he[ph];
        int ws = s_ws[pw], we = s_we[pw];

        const float* __restrict__ base = fb + (size_t)o * (H_ * W_);

        // Prefetch next iteration's channel window while we gather this one
        // (emits global_prefetch_b8 on gfx1250).
        int on = o + blockDim.x;
        if (on < OUTS_PER_ROI) {
            __builtin_prefetch(fb + (size_t)on * (H_ * W_) + (size_t)hs * W_ + ws, 0, 2);
        }

        float sum = 0.0f;
        for (int h = hs; h < he; ++h) {
            const float* __restrict__ row = base + h * W_;
            for (int w = ws; w < we; ++w) {
                sum += row[w];
            }
        }
        int area = (he - hs) * (we - ws);
        ob[o] = (area > 0) ? sum / (float)area : 0.0f;
    }
}

extern "C" void kernel_launch(void* const* d_in, const int* in_sizes, int n_in,
                              void* d_out, int out_size, void* d_ws, size_t ws_size,
                              hipStream_t stream) {
    (void)n_in; (void)out_size; (void)d_ws; (void)ws_size;
    const float* feat = (const float*)d_in[0];
    const float* rois = (const float*)d_in[1];
    float* out        = (float*)d_out;
    const int R = in_sizes[1] / 5;   // rois is (R,5)
    psroi_pool_kernel<<<R, 256, 0, stream>>>(feat, rois, out);
}